// HGAPNet_22935125360767
// MI455X (gfx1250) — compile-verified
//
#include <hip/hip_runtime.h>
#include <hip/hip_bf16.h>
#include <math.h>

// ---------------------------------------------------------------------------
// HGAPNet forward for MI455X (gfx1250).
// * All linear layers on the fp32 tensor path: v_wmma_f32_16x16x4_f32,
//   one wave32 per 16x16 tile, branch-free software-pipelined inner loop.
// * KNN: one wave per query, per-lane private top-k + wave32 shfl_xor merge,
//   global_prefetch_b8 stream-ahead.
// * BN stats: coalesced 16ch x 16row blocks; flag-driven fused elementwise.
// Scratch: deterministic bump arena in d_ws. Params consumed in the
// reference setup_inputs() dict insertion order (97 leaves).
// ---------------------------------------------------------------------------

typedef __attribute__((ext_vector_type(2))) float v2f;
typedef __attribute__((ext_vector_type(8))) float v8f;

#define GF_BIAS 1
#define GF_ADD  2

// Y[M,N] = X[M,K] @ W[K,N] (+ bias[N]) (+ ADD[M,N]).  M % 16 == 0.
__global__ __launch_bounds__(32) void k_gemm(const float* __restrict__ X,
    const float* __restrict__ W, const float* __restrict__ bias,
    const float* __restrict__ ADD, float* __restrict__ Y,
    int M, int K, int N, int flags)
{
  const int lane = threadIdx.x;
  const int half = lane >> 4, l16 = lane & 15;
  const int tm = blockIdx.x, tn = blockIdx.y;
  const int row_a = tm * 16 + l16;
  const int col_b = tn * 16 + l16;
  const bool cb = col_b < N;
  const int colL = cb ? col_b : (N - 1);       // clamped: always in-bounds
  v8f acc = {0.f, 0.f, 0.f, 0.f, 0.f, 0.f, 0.f, 0.f};

  if ((K & 3) == 0) {
    // Fast path: unconditional loads, no exec juggling, 2-stage pipeline.
    // A fragment (16x4 MxK): lane l holds A[m=l&15][k = (l>>4)*2 + {0,1}]
    //   -> one 8-byte vector load (X row offset is 8B aligned since K%4==0).
    // B fragment (4x16 KxN): lane l holds B[k=(l>>4)*2 + {0,1}][n=l&15].
    const float* Xp = X + (long long)row_a * K + half * 2;
    const float* Wp = W + (long long)(half * 2) * N + colL;
    const long long WN = (long long)N;
    v2f a = *(const v2f*)Xp;
    v2f b; b.x = Wp[0]; b.y = Wp[WN];
    for (int kb = 4; kb < K; kb += 4) {
      Xp += 4; Wp += 4 * WN;
      v2f a2 = *(const v2f*)Xp;                 // issue next loads first,
      v2f b2; b2.x = Wp[0]; b2.y = Wp[WN];      // then consume current regs
      v2f bc = b;
      bc.x = cb ? bc.x : 0.f;
      bc.y = cb ? bc.y : 0.f;
      acc = __builtin_amdgcn_wmma_f32_16x16x4_f32(
          false, a, false, bc, (short)0, acc, false, false);
      a = a2; b = b2;
    }
    b.x = cb ? b.x : 0.f;
    b.y = cb ? b.y : 0.f;
    acc = __builtin_amdgcn_wmma_f32_16x16x4_f32(
        false, a, false, b, (short)0, acc, false, false);
  } else {
    // Generic path (K=3 embed): clamped-index unconditional loads + selects.
    for (int kb = 0; kb < K; kb += 4) {
      const int k0 = kb + half * 2, k1 = k0 + 1;
      const int k0c = (k0 < K) ? k0 : (K - 1);
      const int k1c = (k1 < K) ? k1 : (K - 1);
      const float ax = X[(long long)row_a * K + k0c];
      const float ay = X[(long long)row_a * K + k1c];
      const float bx = W[(long long)k0c * N + colL];
      const float by = W[(long long)k1c * N + colL];
      v2f a, b;
      a.x = (k0 < K) ? ax : 0.f;
      a.y = (k1 < K) ? ay : 0.f;
      b.x = (k0 < K && cb) ? bx : 0.f;
      b.y = (k1 < K && cb) ? by : 0.f;
      acc = __builtin_amdgcn_wmma_f32_16x16x4_f32(
          false, a, false, b, (short)0, acc, false, false);
    }
  }

  if (cb) {
    const float bv = (flags & GF_BIAS) ? bias[col_b] : 0.f;
#pragma unroll
    for (int r = 0; r < 8; ++r) {
      const long long row = tm * 16 + half * 8 + r;  // C/D: vgpr r -> m = half*8+r
      float v = acc[r] + bv;
      if (flags & GF_ADD) v += ADD[row * N + col_b];
      Y[row * N + col_b] = v;
    }
  }
}

// Per-channel mean and rsqrt(var + 1e-5) over M rows. N % 16 == 0.
// Block = 16 channels x 16 row-strides -> fully coalesced reads.
__global__ __launch_bounds__(256) void k_colstats(const float* __restrict__ Y,
    int M, int N, float* __restrict__ mean, float* __restrict__ rstd)
{
  const int cl = threadIdx.x & 15;       // channel within group
  const int rt = threadIdx.x >> 4;       // row-stride lane (0..15)
  const int ch = blockIdx.x * 16 + cl;
  float s = 0.f, s2 = 0.f;
  for (int r = rt; r < M; r += 16) {
    const float v = Y[(long long)r * N + ch];
    s += v; s2 += v * v;
  }
  __shared__ float sh[256], sh2[256];
  sh[threadIdx.x] = s; sh2[threadIdx.x] = s2;
  __syncthreads();
  for (int o = 8; o > 0; o >>= 1) {
    if (rt < o) {
      sh[threadIdx.x]  += sh[threadIdx.x + o * 16];
      sh2[threadIdx.x] += sh2[threadIdx.x + o * 16];
    }
    __syncthreads();
  }
  if (rt == 0) {
    const float m = sh[cl] / (float)M;
    float var = sh2[cl] / (float)M - m * m;
    if (var < 0.f) var = 0.f;
    mean[ch] = m;
    rstd[ch] = rsqrtf(var + 1e-5f);
  }
}

#define BF_SUBMEAN 1
#define BF_RELU    2
#define BF_RESADD  4
#define BF_AFFINE  8

__global__ void k_bn_act(float* __restrict__ Y, const float* __restrict__ mean,
    const float* __restrict__ rstd, const float* __restrict__ g,
    const float* __restrict__ bb, const float* __restrict__ res,
    long long total, int N, int flags)
{
  const long long i = (long long)blockIdx.x * blockDim.x + threadIdx.x;
  if (i >= total) return;
  const int ch = (int)(i % N);
  float v = Y[i];
  if (flags & BF_SUBMEAN) v -= mean[ch];
  v *= rstd[ch];
  if (flags & BF_AFFINE) v = v * g[ch] + bb[ch];
  if (flags & BF_RESADD) v += res[i];
  if (flags & BF_RELU)   v = fmaxf(v, 0.f);
  Y[i] = v;
}

// Exact k-NN: one wave32 per query. Private top-k per lane, shfl merge.
#define MAXK 17
__global__ __launch_bounds__(32) void k_knn(const float* __restrict__ Q,
    const float* __restrict__ P, int B, int Mq, int Nk, int k,
    int* __restrict__ oidx, float* __restrict__ od2)
{
  const int q = blockIdx.x;              // 0 .. B*Mq-1
  const int b = q / Mq;
  const float* Pb = P + (long long)b * Nk * 3;
  const float qx = Q[(long long)q * 3 + 0];
  const float qy = Q[(long long)q * 3 + 1];
  const float qz = Q[(long long)q * 3 + 2];
  const int lane = threadIdx.x;

  float bd[MAXK]; int bi[MAXK];
  int cnt = 0, wpos = 0; float worst = -1.f;
  for (int j = lane; j < Nk; j += 32) {
    __builtin_prefetch(Pb + (long long)(j + 256) * 3, 0, 0);  // global_prefetch_b8
    const float dx = Pb[(long long)j * 3 + 0] - qx;
    const float dy = Pb[(long long)j * 3 + 1] - qy;
    const float dz = Pb[(long long)j * 3 + 2] - qz;
    const float d = dx * dx + dy * dy + dz * dz;
    if (cnt < k) {
      bd[cnt] = d; bi[cnt] = j;
      if (cnt == 0 || d > worst) { worst = d; wpos = cnt; }
      cnt++;
    } else if (d < worst) {
      bd[wpos] = d; bi[wpos] = j;
      worst = bd[0]; wpos = 0;
      for (int t = 1; t < k; ++t) if (bd[t] > worst) { worst = bd[t]; wpos = t; }
    }
  }
  // sort private list ascending
  for (int a = 0; a < cnt; ++a) {
    int mb = a;
    for (int t = a + 1; t < cnt; ++t) if (bd[t] < bd[mb]) mb = t;
    float td = bd[a]; bd[a] = bd[mb]; bd[mb] = td;
    int ti = bi[a]; bi[a] = bi[mb]; bi[mb] = ti;
  }
  // wave-wide k-way merge
  int ptr = 0;
  for (int s = 0; s < k; ++s) {
    const float cv = (ptr < cnt) ? bd[ptr] : 3.4e38f;
    const int   ci = (ptr < cnt) ? bi[ptr] : 0;
    float v = cv; int wl = lane;
    for (int off = 16; off > 0; off >>= 1) {
      const float ov = __shfl_xor(v, off, 32);
      const int owl = __shfl_xor(wl, off, 32);
      if (ov < v || (ov == v && owl < wl)) { v = ov; wl = owl; }
    }
    const float wd = __shfl(cv, wl, 32);
    const int   wi = __shfl(ci, wl, 32);
    if (lane == wl) ptr++;
    if (lane == 0) { oidx[(long long)q * k + s] = wi;
                     od2[(long long)q * k + s] = wd; }
  }
}

// Y[b, m, c] = X[b, m, c] for m < Mout (strided prefix pooling copy)
__global__ void k_pool(const float* __restrict__ X, float* __restrict__ Y,
                       int B, int Nin, int Mout, int C)
{
  const long long total = (long long)B * Mout * C;
  const long long i = (long long)blockIdx.x * blockDim.x + threadIdx.x;
  if (i >= total) return;
  const long long per = (long long)Mout * C;
  const int b = (int)(i / per);
  const long long rem = i % per;
  Y[i] = X[(long long)b * Nin * C + rem];
}

// Y[row, c] = X[b(row), idx[row], c]
__global__ void k_gather(const float* __restrict__ X, const int* __restrict__ idx,
                         float* __restrict__ Y, int B, int Mq, int k, int C, int Nin)
{
  const long long total = (long long)B * Mq * k * C;
  const long long i = (long long)blockIdx.x * blockDim.x + threadIdx.x;
  if (i >= total) return;
  const long long row = i / C;
  const int c = (int)(i % C);
  const int b = (int)(row / ((long long)Mq * k));
  const int id = idx[row];
  Y[i] = X[((long long)b * Nin + id) * C + c];
}

// pein[row, 0..3] = {rel.xyz, |rel|}
__global__ void k_pein(const float* __restrict__ pos, const float* __restrict__ posq,
                       const int* __restrict__ idx, float* __restrict__ pein,
                       int B, int Mq, int k, int Nin)
{
  const long long total = (long long)B * Mq * k;
  const long long row = (long long)blockIdx.x * blockDim.x + threadIdx.x;
  if (row >= total) return;
  const int b = (int)(row / ((long long)Mq * k));
  const int m = (int)((row / k) % Mq);
  const int id = idx[row];
  const long long pb = ((long long)b * Nin + id) * 3;
  const long long qb = ((long long)b * Mq + m) * 3;
  const float rx = pos[pb + 0] - posq[qb + 0];
  const float ry = pos[pb + 1] - posq[qb + 1];
  const float rz = pos[pb + 2] - posq[qb + 2];
  pein[row * 4 + 0] = rx;
  pein[row * 4 + 1] = ry;
  pein[row * 4 + 2] = rz;
  pein[row * 4 + 3] = sqrtf(rx * rx + ry * ry + rz * rz);
}

// F[bm, j, c] -= XC[bm, c]  (in-place)
__global__ void k_subcenter(float* __restrict__ F, const float* __restrict__ XC,
                            long long total, int C, int k)
{
  const long long i = (long long)blockIdx.x * blockDim.x + threadIdx.x;
  if (i >= total) return;
  const int c = (int)(i % C);
  const long long bm = (i / C) / k;
  F[i] -= XC[bm * C + c];
}

// H[bm, c] = sum_j softmax_j(L[bm, j, c]) * F[bm, j, c]
__global__ void k_attn_pool(const float* __restrict__ L, const float* __restrict__ F,
                            float* __restrict__ H, long long BM, int k, int C)
{
  const long long total = BM * C;
  const long long i = (long long)blockIdx.x * blockDim.x + threadIdx.x;
  if (i >= total) return;
  const long long bm = i / C;
  const int c = (int)(i % C);
  const float* lp = L + bm * k * C + c;
  const float* fp = F + bm * k * C + c;
  float mx = -3.4e38f;
  for (int j = 0; j < k; ++j) mx = fmaxf(mx, lp[(long long)j * C]);
  float se = 0.f, acc = 0.f;
  for (int j = 0; j < k; ++j) {
    const float e = expf(lp[(long long)j * C] - mx);
    se += e;
    acc += e * fp[(long long)j * C];
  }
  H[i] = acc / se;
}

// Inverse-distance 3-NN interpolation of coarse features to fine points.
__global__ void k_interp(const float* __restrict__ XC, const int* __restrict__ idx,
                         const float* __restrict__ d2, float* __restrict__ Y,
                         int B, int Mf, int Nc, int C)
{
  const long long total = (long long)B * Mf * C;
  const long long i = (long long)blockIdx.x * blockDim.x + threadIdx.x;
  if (i >= total) return;
  const long long row = i / C;
  const int c = (int)(i % C);
  const int b = (int)(row / Mf);
  float ws = 0.f, acc = 0.f;
#pragma unroll
  for (int j = 0; j < 3; ++j) {
    const float w = 1.f / fmaxf(d2[row * 3 + j], 1e-16f);
    const int id = idx[row * 3 + j];
    ws += w;
    acc += w * XC[((long long)b * Nc + id) * C + c];
  }
  Y[i] = acc / ws;
}

// Y[r, :] = concat(A[r, 0..c1), Bm[r, 0..c2))
__global__ void k_concat(const float* __restrict__ A, const float* __restrict__ Bm,
                         float* __restrict__ Y, long long Mrows, int c1, int c2)
{
  const int ct = c1 + c2;
  const long long total = Mrows * ct;
  const long long i = (long long)blockIdx.x * blockDim.x + threadIdx.x;
  if (i >= total) return;
  const long long r = i / ct;
  const int c = (int)(i % ct);
  Y[i] = (c < c1) ? A[r * c1 + c] : Bm[r * c2 + (c - c1)];
}

// --------------------------- host orchestration ----------------------------

namespace {
struct Lin { const float* W; const float* b; };

inline void gemm(const float* X, const float* W, const float* b, const float* ADD,
                 float* Y, int M, int K, int N, int flags, hipStream_t s)
{
  dim3 g(M / 16, (N + 15) / 16);
  k_gemm<<<g, 32, 0, s>>>(X, W, b, ADD, Y, M, K, N, flags);
}
inline void stats(const float* Y, int M, int N, float* mean, float* rstd, hipStream_t s)
{
  k_colstats<<<N / 16, 256, 0, s>>>(Y, M, N, mean, rstd);
}
inline void bnact(float* Y, const float* mean, const float* rstd, const float* g,
                  const float* bb, const float* res, long long total, int N,
                  int flags, hipStream_t s)
{
  const int nb = (int)((total + 255) / 256);
  k_bn_act<<<nb, 256, 0, s>>>(Y, mean, rstd, g, bb, res, total, N, flags);
}
// y = relu(bn(X @ W + b))
inline void mlp(const float* X, Lin p, float* Y, int M, int K, int N,
                float* mean, float* rstd, hipStream_t s)
{
  gemm(X, p.W, p.b, nullptr, Y, M, K, N, GF_BIAS, s);
  stats(Y, M, N, mean, rstd, s);
  bnact(Y, mean, rstd, nullptr, nullptr, nullptr, (long long)M * N, N,
        BF_SUBMEAN | BF_RELU, s);
}
// OUT = relu(bn(relu(bn(X@l1))@l2) + X)
inline void resmlp(const float* X, Lin l1, Lin l2, float* T1, float* OUT,
                   int M, int C, float* mean, float* rstd, hipStream_t s)
{
  gemm(X, l1.W, l1.b, nullptr, T1, M, C, C, GF_BIAS, s);
  stats(T1, M, C, mean, rstd, s);
  bnact(T1, mean, rstd, nullptr, nullptr, nullptr, (long long)M * C, C,
        BF_SUBMEAN | BF_RELU, s);
  gemm(T1, l2.W, l2.b, nullptr, OUT, M, C, C, GF_BIAS, s);
  stats(OUT, M, C, mean, rstd, s);
  bnact(OUT, mean, rstd, nullptr, nullptr, X, (long long)M * C, C,
        BF_SUBMEAN | BF_RESADD | BF_RELU, s);
}
inline int eblocks(long long total) { return (int)((total + 255) / 256); }
} // namespace

extern "C" void kernel_launch(void* const* d_in, const int* in_sizes, int n_in,
                              void* d_out, int out_size, void* d_ws, size_t ws_size,
                              hipStream_t stream)
{
  (void)in_sizes; (void)n_in; (void)out_size; (void)ws_size;
  const int B = 2, N0 = 16384, NB = 17, IK = 3;
  const int pts[5] = {16384, 4096, 1024, 256, 64};
  const int ch[5]  = {64, 128, 256, 512, 1024};

  // ---- parameters: reference dict insertion-order flatten (97 leaves) ----
  int t = 0;
  const float* pos = (const float*)d_in[t++];
  auto nxt = [&]() -> const float* { return (const float*)d_in[t++]; };
  auto lin = [&]() -> Lin { Lin l; l.W = nxt(); l.b = nxt(); return l; };

  Lin embed = lin();
  struct EncP { Lin pe, attn; const float *g, *bb; Lin g1, g2, rl, r1, r2; } enc[4];
  for (int i = 0; i < 4; ++i) {
    enc[i].pe = lin(); enc[i].attn = lin();
    enc[i].g = nxt(); enc[i].bb = nxt();
    enc[i].g1 = lin(); enc[i].g2 = lin();
    enc[i].rl = lin();
    enc[i].r1 = lin(); enc[i].r2 = lin();
  }
  struct DecP { Lin l, r1, r2; } dec[4];
  for (int i = 0; i < 4; ++i) { dec[i].l = lin(); dec[i].r1 = lin(); dec[i].r2 = lin(); }
  Lin h1 = lin(), h2 = lin(), h3 = lin();

  // ---- workspace arena (deterministic bump) ----
  char* base = (char*)d_ws;
  size_t off = 0;
  auto alloc = [&](size_t nfloats) -> float* {
    float* p = (float*)(base + off);
    off += ((nfloats * sizeof(float) + 255) / 256) * 256;
    return p;
  };

  const float* posL[5]; posL[0] = pos;
  float* posW[5] = {nullptr, nullptr, nullptr, nullptr, nullptr};
  for (int l = 1; l < 5; ++l) { posW[l] = alloc((size_t)B * pts[l] * 3); posL[l] = posW[l]; }
  float* xL[5];
  for (int l = 0; l < 5; ++l) xL[l] = alloc((size_t)B * pts[l] * ch[l]);
  float* mean = alloc(2048);
  float* rstd = alloc(2048);
  int*   S_IDX = (int*)alloc(139264);
  float* S_D2  = alloc(139264);
  float* S_PE  = alloc(557056);
  float* SA = alloc(8912896);   // neighbor-feature / temp slot
  float* SB = alloc(8912896);   // f / d / concat slot
  float* SC = alloc(8912896);   // logits / temp slot
  float* SD = alloc(4194304);   // x_pool / res_lin-out slot
  float* SE = alloc(524288);    // attention-pooled h slot
  float* XIA = alloc(2097152);  // decoder ping
  float* XIB = alloc(2097152);  // decoder pong

  // ---- embed: x0 = relu(bn(pos @ We + be)) ----
  mlp(pos, embed, xL[0], B * N0, 3, 64, mean, rstd, stream);

  // ---- encoder ----
  for (int i = 0; i < 4; ++i) {
    const int c = ch[i], Nc = pts[i], M = pts[i + 1];
    const int rows = B * M;
    const int r17 = rows * NB;

    k_pool<<<eblocks((long long)rows * 3), 256, 0, stream>>>(posL[i], posW[i + 1], B, Nc, M, 3);
    k_pool<<<eblocks((long long)rows * c), 256, 0, stream>>>(xL[i], SD, B, Nc, M, c);
    k_knn<<<rows, 32, 0, stream>>>(posL[i + 1], posL[i], B, M, Nc, NB, S_IDX, S_D2);
    k_gather<<<eblocks((long long)r17 * c), 256, 0, stream>>>(xL[i], S_IDX, SA, B, M, NB, c, Nc);
    k_pein<<<eblocks((long long)r17), 256, 0, stream>>>(posL[i], posL[i + 1], S_IDX, S_PE, B, M, NB, Nc);
    // f = pein @ peW + peb + nb_x
    gemm(S_PE, enc[i].pe.W, enc[i].pe.b, SA, SB, r17, 4, c, GF_BIAS | GF_ADD, stream);
    // d = f - x_pool ; f = d * rstd(d) * g + b   (no mean subtraction)
    k_subcenter<<<eblocks((long long)r17 * c), 256, 0, stream>>>(SB, SD, (long long)r17 * c, c, NB);
    stats(SB, r17, c, mean, rstd, stream);
    bnact(SB, mean, rstd, enc[i].g, enc[i].bb, nullptr, (long long)r17 * c, c, BF_AFFINE, stream);
    // logits = f @ Wa + ba ; h = sum_j softmax_j(logits) * f
    gemm(SB, enc[i].attn.W, enc[i].attn.b, nullptr, SC, r17, c, c, GF_BIAS, stream);
    k_attn_pool<<<eblocks((long long)rows * c), 256, 0, stream>>>(SC, SB, SE, (long long)rows, NB, c);
    // gapl residual block (c), channel-doubling mlp, residual block (2c)
    resmlp(SE, enc[i].g1, enc[i].g2, SA, SC, rows, c, mean, rstd, stream);
    mlp(SC, enc[i].rl, SD, rows, c, 2 * c, mean, rstd, stream);
    resmlp(SD, enc[i].r1, enc[i].r2, SA, xL[i + 1], rows, 2 * c, mean, rstd, stream);
  }

  // ---- decoder ----
  float* xi = xL[4];
  float* outs[4] = {XIA, XIB, XIA, XIB};
  for (int i = 0; i < 4; ++i) {
    const int Cc = ch[4 - i], Cs = ch[3 - i];
    const int Mc = pts[4 - i], Mf = pts[3 - i];
    const int rows = B * Mf;
    k_knn<<<rows, 32, 0, stream>>>(posL[3 - i], posL[4 - i], B, Mf, Mc, IK, S_IDX, S_D2);
    k_interp<<<eblocks((long long)rows * Cc), 256, 0, stream>>>(xi, S_IDX, S_D2, SA, B, Mf, Mc, Cc);
    k_concat<<<eblocks((long long)rows * (Cs + Cc)), 256, 0, stream>>>(xL[3 - i], SA, SB, (long long)rows, Cs, Cc);
    mlp(SB, dec[i].l, SC, rows, Cs + Cc, Cs, mean, rstd, stream);
    resmlp(SC, dec[i].r1, dec[i].r2, SA, outs[i], rows, Cs, mean, rstd, stream);
    xi = outs[i];
  }

  // ---- head: 64 -> 128 -> 128 -> 13 ----
  mlp(xi, h1, SA, B * N0, 64, 128, mean, rstd, stream);
  mlp(SA, h2, SB, B * N0, 128, 128, mean, rstd, stream);
  gemm(SB, h3.W, h3.b, nullptr, (float*)d_out, B * N0, 128, 13, GF_BIAS, stream);
}